// AdversMaskEdge_13048110645526
// MI455X (gfx1250) — compile-verified
//
#include <hip/hip_runtime.h>
#include <hip/hip_bf16.h>
#include <math.h>

typedef __attribute__((ext_vector_type(2))) float v2f;
typedef __attribute__((ext_vector_type(8))) float v8f;

#define LL 2
#define NNODE 10000
#define DD 128
// K-pair-major LDS layout: element (k, n) lives at (k>>1)*KPSTRIDE + 2n + (k&1).
// KPSTRIDE = 160 floats: 160 mod 64 = 32 -> the two wave-halves (h2) hit
// disjoint 32-bank sets on b64 fragment reads -> conflict-free.
#define KPSTRIDE 160

__global__ __launch_bounds__(256) void advers_mask_edge_kernel(
    const float* __restrict__ h,    // (2, N, 128)
    const float* __restrict__ W0,   // (128, 512)
    const float* __restrict__ b0,   // (128,)
    const float* __restrict__ W1,   // (128, 128)
    const float* __restrict__ b1,   // (128,)
    const float* __restrict__ Wf,   // (2, 128)
    const float* __restrict__ bf,   // (2,)
    const float* __restrict__ u,    // (E, 2)
    const int*   __restrict__ src,  // (E,)
    const int*   __restrict__ dst,  // (E,)
    float* __restrict__ out,        // (E, 2)
    int Etot)
{
    __shared__ float Xs[64 * KPSTRIDE];   // 40 KB: X chunk / hidden / pos (reused)
    __shared__ int   sidx[64];
    __shared__ int   didx[64];
    __shared__ float Sc[2][64];

    const int t      = threadIdx.x;
    const int lane   = t & 31;
    const int w      = t >> 5;        // wave id 0..7 -> M rows [16w, 16w+16)
    const int n16    = lane & 15;     // A: M row; B/C/D: N col
    const int h2     = lane >> 4;     // half
    const int e_base = blockIdx.x * 64;

    // L2 warm hint for resident weights (global_prefetch_b8)
    __builtin_prefetch(W0 + t * 256, 0, 3);
    __builtin_prefetch(W1 + t * 64, 0, 3);

    // ---- load edge indices for this tile (clamped for tail safety) ----
    if (t < 64) {
        int eg = e_base + t; if (eg >= Etot) eg = Etot - 1;
        sidx[t] = src[eg];
    } else if (t < 128) {
        int eg = e_base + (t - 64); if (eg >= Etot) eg = Etot - 1;
        didx[t - 64] = dst[eg];
    }

    // ---- GEMM1: hidden(128 x 64) = W0(128x512) @ X(512x64) ----
    v8f acc[4] = {};
    const int m = 16 * w + n16;                   // A-matrix row for this lane
    const float* W0p = W0 + m * 512;
    // per-lane base (in floats) of its B-fragment column group: 32j added later
    const int bcol = 2 * n16;

    for (int p = 0; p < 4; ++p) {
        __syncthreads();                          // prev chunk fully consumed
        // stage X chunk for pair p=(l1,l2): X[c][e] = h[l1,src[e],c]*h[l2,dst[e],c]
        {
            const int l1 = p >> 1, l2 = p & 1;
            const int e = t & 63, q = t >> 6;     // 4 thread-groups x 64 edges
            const float* hs = h + ((size_t)l1 * NNODE + sidx[e]) * DD + q * 32;
            const float* hd = h + ((size_t)l2 * NNODE + didx[e]) * DD + q * 32;
            #pragma unroll
            for (int i = 0; i < 32; i += 4) {
                float4 a4 = *(const float4*)(hs + i);
                float4 b4 = *(const float4*)(hd + i);
                const int c  = q * 32 + i;        // even, multiple of 4
                const int kp = c >> 1;
                float2 lo = { a4.x * b4.x, a4.y * b4.y };
                float2 hi = { a4.z * b4.z, a4.w * b4.w };
                *(float2*)&Xs[(kp    ) * KPSTRIDE + 2 * e] = lo;
                *(float2*)&Xs[(kp + 1) * KPSTRIDE + 2 * e] = hi;
            }
        }
        __syncthreads();
        // consume chunk: 32 K-steps x 4 N-tiles of v_wmma_f32_16x16x4_f32
        for (int kb = 0; kb < 32; ++kb) {
            const int k2 = 4 * kb + 2 * h2;       // lane's K pair (even)
            float2 av = *(const float2*)(W0p + p * 128 + k2);
            v2f A = {av.x, av.y};
            const float* brow = &Xs[(2 * kb + h2) * KPSTRIDE + bcol];
            #pragma unroll
            for (int j = 0; j < 4; ++j) {
                v2f B = *(const v2f*)(brow + 32 * j);   // single ds_load_b64
                acc[j] = __builtin_amdgcn_wmma_f32_16x16x4_f32(
                    false, A, false, B, (short)0, acc[j], false, false);
            }
        }
    }

    // ---- bias + ReLU, spill hidden to LDS (reuse Xs, same pair layout) ----
    __syncthreads();
    {
        float b0v[8];
        #pragma unroll
        for (int r = 0; r < 8; ++r) b0v[r] = b0[16 * w + 8 * h2 + r];
        #pragma unroll
        for (int j = 0; j < 4; ++j)
            #pragma unroll
            for (int r = 0; r < 8; r += 2) {
                // rows m0=16w+8h2+r (even) and m0+1 are LDS-adjacent
                float2 hv = { fmaxf(acc[j][r]     + b0v[r],     0.0f),
                              fmaxf(acc[j][r + 1] + b0v[r + 1], 0.0f) };
                const int mp = 8 * w + 4 * h2 + (r >> 1);
                *(float2*)&Xs[mp * KPSTRIDE + 32 * j + 2 * n16] = hv;
            }
    }
    __syncthreads();

    // ---- GEMM2: pos(128 x 64) = W1(128x128) @ hidden(128x64) ----
    v8f acc2[4] = {};
    const float* W1p = W1 + m * 128;
    for (int kb = 0; kb < 32; ++kb) {
        const int k2 = 4 * kb + 2 * h2;
        float2 av = *(const float2*)(W1p + k2);
        v2f A = {av.x, av.y};
        const float* brow = &Xs[(2 * kb + h2) * KPSTRIDE + bcol];
        #pragma unroll
        for (int j = 0; j < 4; ++j) {
            v2f B = *(const v2f*)(brow + 32 * j);
            acc2[j] = __builtin_amdgcn_wmma_f32_16x16x4_f32(
                false, A, false, B, (short)0, acc2[j], false, false);
        }
    }
    __syncthreads();
    {
        float b1v[8];
        #pragma unroll
        for (int r = 0; r < 8; ++r) b1v[r] = b1[16 * w + 8 * h2 + r];
        #pragma unroll
        for (int j = 0; j < 4; ++j)
            #pragma unroll
            for (int r = 0; r < 8; r += 2) {
                float2 pv = { acc2[j][r]     + b1v[r],
                              acc2[j][r + 1] + b1v[r + 1] };
                const int mp = 8 * w + 4 * h2 + (r >> 1);
                *(float2*)&Xs[mp * KPSTRIDE + 32 * j + 2 * n16] = pv;
            }
    }
    __syncthreads();

    // ---- logits + Gumbel + hard one-hot (z == y_hard exactly in fwd) ----
    if (t < 128) {
        const int e   = t & 63;
        const int cls = t >> 6;
        const float* Wfp = Wf + cls * 128;
        float acc3 = bf[cls];
        #pragma unroll 8
        for (int c = 0; c < 128; ++c)
            acc3 += Wfp[c] * Xs[(c >> 1) * KPSTRIDE + 2 * e + (c & 1)];
        int eg = e_base + e;
        float uval = (eg < Etot) ? u[eg * 2 + cls] : 0.5f;
        float g = -logf(-logf(uval + 1e-10f) + 1e-10f);
        Sc[cls][e] = acc3 + g;
    }
    __syncthreads();
    if (t < 128) {
        const int e   = t & 63;
        const int cls = t >> 6;
        int eg = e_base + e;
        if (eg < Etot) {
            float s0 = Sc[0][e], s1 = Sc[1][e];
            // jnp.argmax: first max wins ties -> class 0 on tie
            float v = (cls == 0) ? ((s0 >= s1) ? 1.0f : 0.0f)
                                 : ((s1 > s0) ? 1.0f : 0.0f);
            out[eg * 2 + cls] = v;
        }
    }
}

extern "C" void kernel_launch(void* const* d_in, const int* in_sizes, int n_in,
                              void* d_out, int out_size, void* d_ws, size_t ws_size,
                              hipStream_t stream) {
    const float* h  = (const float*)d_in[0];
    const float* W0 = (const float*)d_in[1];
    const float* b0 = (const float*)d_in[2];
    const float* W1 = (const float*)d_in[3];
    const float* b1 = (const float*)d_in[4];
    const float* Wf = (const float*)d_in[5];
    const float* bf = (const float*)d_in[6];
    const float* u  = (const float*)d_in[7];
    const int* src  = (const int*)d_in[8];
    const int* dst  = (const int*)d_in[9];
    float* out = (float*)d_out;
    const int E = in_sizes[8];

    dim3 grid((E + 63) / 64), block(256);
    advers_mask_edge_kernel<<<grid, block, 0, stream>>>(
        h, W0, b0, W1, b1, Wf, bf, u, src, dst, out, E);
}